// Bilinear_14104672600541
// MI455X (gfx1250) — compile-verified
//
#include <hip/hip_runtime.h>
#include <hip/hip_bf16.h>

// Bilinear: out[n,o] = sum_{i,j} x1[n,i] * W[o,i,j] * x2[n,j] + bias[o]
// N=8192, I=J=256, O=256.
// Single fused GEMM: out = Wflat(O x 65536) * P(65536 x N),
// P[k,n] = x1[n, k/256] * x2[n, k%256] generated in registers.
// W (64MB) lives in the 192MB L2; x1/x2 node tiles live in LDS.

#define NODE 8192
#define IN1  256
#define IN2  256
#define OUTD 256

typedef __attribute__((ext_vector_type(8))) float v8f;
typedef __attribute__((ext_vector_type(2))) float v2f;
typedef __attribute__((ext_vector_type(4))) float v4f;

// 256 threads = 8 waves (wave32). Block covers 32 nodes x all 256 outputs.
// Wave w handles o-tiles {2w, 2w+1}; each with 2 node-subtiles of 16.
__global__ __launch_bounds__(256) void
bilinear_wmma_f32(const float* __restrict__ x1,
                  const float* __restrict__ x2,
                  const float* __restrict__ W,
                  const float* __restrict__ bias,
                  float* __restrict__ out)
{
    __shared__ float s_x1[32 * IN1];   // 32 KB
    __shared__ float s_x2[32 * IN2];   // 32 KB

    const int tid  = threadIdx.x;
    const int wave = tid >> 5;         // 0..7
    const int lane = tid & 31;
    const int lm   = lane & 15;        // row within half-wave
    const int koff = (lane >> 4) << 1; // lanes 0-15 -> K0,K1 ; lanes 16-31 -> K2,K3
    const int nb   = blockIdx.x;       // node block (32 nodes)

    // ---- Stage x1/x2 tiles (32 x 256 each) into LDS with b128 loads ----
    const size_t nbase = (size_t)nb * 32;
    #pragma unroll
    for (int k = 0; k < 8; ++k) {
        int idx = tid + k * 256;            // float4 index, 0..2047
        int row = idx >> 6;                 // 0..31
        int c4  = (idx & 63) << 2;          // 0..252
        *(v4f*)&s_x1[row * IN1 + c4] =
            *(const v4f*)&x1[(nbase + row) * IN1 + c4];
        *(v4f*)&s_x2[row * IN2 + c4] =
            *(const v4f*)&x2[(nbase + row) * IN2 + c4];
    }
    __syncthreads();

    // 4 independent accumulators: [o-tile 0/1][node-subtile 0/1]
    v8f acc00 = {}, acc01 = {}, acc10 = {}, acc11 = {};

    // A-operand base pointers: lane lm supplies W row o = ot*16 + lm,
    // K-halves split by koff (2 consecutive floats per lane -> b64 load).
    const float* w0 = W + ((size_t)(wave * 2 + 0) * 16 + lm) * (IN1 * IN2) + koff;
    const float* w1 = W + ((size_t)(wave * 2 + 1) * 16 + lm) * (IN1 * IN2) + koff;

    const float* x2r0 = &s_x2[(size_t)lm * IN2 + koff];        // node-subtile 0
    const float* x2r1 = &s_x2[(size_t)(16 + lm) * IN2 + koff]; // node-subtile 1

    for (int i = 0; i < IN1; ++i) {
        // x1[n,i] is invariant over the j loop: hoist per node-subtile.
        const float a0 = s_x1[(size_t)lm * IN1 + i];
        const float a1 = s_x1[(size_t)(16 + lm) * IN1 + i];
        const int kbase = i * IN2;

        #pragma unroll 4
        for (int jb = 0; jb < IN2; jb += 4) {
            // B-operand: P[k,n] = x1[n,i] * x2[n,j]  (2 floats per lane)
            v2f xb0 = *(const v2f*)&x2r0[jb];   // ds_load_b64
            v2f xb1 = *(const v2f*)&x2r1[jb];
            v2f b0 = xb0 * a0;
            v2f b1 = xb1 * a1;
            // A-operand: W rows, contiguous k (L2-resident) -> global b64
            v2f A0 = *(const v2f*)&w0[kbase + jb];
            v2f A1 = *(const v2f*)&w1[kbase + jb];

            acc00 = __builtin_amdgcn_wmma_f32_16x16x4_f32(
                        false, A0, false, b0, (short)0, acc00, false, false);
            acc01 = __builtin_amdgcn_wmma_f32_16x16x4_f32(
                        false, A0, false, b1, (short)0, acc01, false, false);
            acc10 = __builtin_amdgcn_wmma_f32_16x16x4_f32(
                        false, A1, false, b0, (short)0, acc10, false, false);
            acc11 = __builtin_amdgcn_wmma_f32_16x16x4_f32(
                        false, A1, false, b1, (short)0, acc11, false, false);
        }
    }

    // ---- Epilogue: D layout is [M = r + 8*(lane>=16)][N = lane&15] ----
    const int mofs = (lane >> 4) << 3;   // 0 or 8
    v8f accs[2][2] = {{acc00, acc01}, {acc10, acc11}};
    #pragma unroll
    for (int ot = 0; ot < 2; ++ot) {
        const int obase = (wave * 2 + ot) * 16 + mofs;
        #pragma unroll
        for (int s = 0; s < 2; ++s) {
            const size_t nrow = nbase + s * 16 + lm;  // N column = node index
            #pragma unroll
            for (int r = 0; r < 8; ++r) {
                const int o = obase + r;
                out[nrow * OUTD + o] = accs[ot][s][r] + bias[o];
            }
        }
    }
}

extern "C" void kernel_launch(void* const* d_in, const int* in_sizes, int n_in,
                              void* d_out, int out_size, void* d_ws, size_t ws_size,
                              hipStream_t stream) {
    const float* x1   = (const float*)d_in[0];
    const float* x2   = (const float*)d_in[1];
    const float* W    = (const float*)d_in[2];
    const float* bias = (const float*)d_in[3];
    float* out = (float*)d_out;

    dim3 grid(NODE / 32);   // 256 blocks, each: 32 nodes x 256 outputs
    dim3 block(256);        // 8 waves of 32
    bilinear_wmma_f32<<<grid, block, 0, stream>>>(x1, x2, W, bias, out);
}